// FFF_80985903333528
// MI455X (gfx1250) — compile-verified
//
#include <hip/hip_runtime.h>
#include <math.h>

#define B_SAMP   8192
#define D_IN     2048
#define D_OUT    2048
#define N_NODES  4095
#define NSTEPS   12      // DEPTH+1 routing steps
#define TOPN     16      // padded top-of-tree nodes (valid ids 0..14)

typedef float v2f __attribute__((ext_vector_type(2)));
typedef float v4f __attribute__((ext_vector_type(4)));
typedef float v8f __attribute__((ext_vector_type(8)));

__device__ __forceinline__ float gelu_exact(float s) {
    // exact GELU: 0.5*s*(1+erf(s/sqrt(2)))
    return 0.5f * s * (1.0f + erff(s * 0.70710678118654752f));
}

// ---------------------------------------------------------------------------
// Kernel 1: transpose W_out (D_OUT x N_NODES) -> WoutT (N_NODES x D_OUT)
// so per-node output columns become contiguous 8KB rows for coalesced gathers.
// ---------------------------------------------------------------------------
__global__ __launch_bounds__(256)
void k_transpose_wout(const float* __restrict__ Wout, float* __restrict__ WoutT) {
    __shared__ float tile[32][33];
    int tx = threadIdx.x, ty = threadIdx.y;   // block = (32, 8)
    int nbase = blockIdx.x * 32;              // node dim (0..4094)
    int obase = blockIdx.y * 32;              // dout dim (0..2047)
    #pragma unroll
    for (int j = 0; j < 4; ++j) {
        int o = obase + ty + j * 8;           // always < D_OUT
        int n = nbase + tx;
        if (n < N_NODES)
            tile[ty + j * 8][tx] = Wout[o * N_NODES + n];
    }
    __syncthreads();
    #pragma unroll
    for (int j = 0; j < 4; ++j) {
        int n = nbase + ty + j * 8;
        int o = obase + tx;
        if (n < N_NODES)
            WoutT[n * D_OUT + o] = tile[tx][ty + j * 8];
    }
}

// ---------------------------------------------------------------------------
// Kernel 2: tree routing. One wave32 per sample; x row held in 64 VGPRs/lane.
// 12 sequential fp32 dots against data-dependent W_in rows. Both children of
// the current node (adjacent rows -> one 16KB span) are prefetched while the
// current dot is computed, hiding one L2 latency per level of the chain.
// x is streamed with non-temporal loads so it doesn't evict W_in/WoutT in L2.
// ---------------------------------------------------------------------------
__global__ __launch_bounds__(256)
void k_route(const float* __restrict__ x, const float* __restrict__ Win,
             int* __restrict__ nodes, float* __restrict__ gvals) {
    int wid  = threadIdx.x >> 5;
    int lane = threadIdx.x & 31;
    int s = blockIdx.x * 8 + wid;             // 1024 blocks * 8 waves = 8192 samples

    const v4f* xp = (const v4f*)(x + (size_t)s * D_IN);
    v4f xv[16];                               // 2048 floats / 32 lanes = 16 float4
    #pragma unroll
    for (int i = 0; i < 16; ++i) xv[i] = __builtin_nontemporal_load(xp + lane + i * 32);

    int cur = 0;
    for (int d = 0; d < NSTEPS; ++d) {
        // Speculatively prefetch both children rows (contiguous 16KB region,
        // 128 lines of 128B) while the current dot product is in flight.
        if (d < NSTEPS - 1) {
            const float* kid = Win + (2 * cur + 1) * D_IN;
            #pragma unroll
            for (int j = 0; j < 4; ++j)
                __builtin_prefetch(kid + (j * 32 + lane) * 32, 0, 3);
        }
        const v4f* wp = (const v4f*)(Win + cur * D_IN);
        float acc = 0.f;
        #pragma unroll
        for (int i = 0; i < 16; ++i) {
            v4f w = wp[lane + i * 32];
            acc = fmaf(xv[i].x, w.x, acc);
            acc = fmaf(xv[i].y, w.y, acc);
            acc = fmaf(xv[i].z, w.z, acc);
            acc = fmaf(xv[i].w, w.w, acc);
        }
        // wave32 butterfly reduction -> every lane holds the full dot product
        #pragma unroll
        for (int off = 16; off > 0; off >>= 1)
            acc += __shfl_xor(acc, off);

        if (lane == 0) {
            nodes[s * NSTEPS + d] = cur;
            gvals[s * NSTEPS + d] = gelu_exact(acc);
        }
        cur = 2 * cur + 1 + (acc >= 0.f ? 1 : 0);
    }
}

// ---------------------------------------------------------------------------
// Kernel 3: conditional output projection for 16 samples per block.
//   Depths 0..3 visit only nodes 0..14 for every sample  -> dense shared
//   contraction  Y += A(16x16) x WoutT[0:16,:]  via fp32 WMMA.
//   Depths 4..11 are divergent -> coalesced row gathers accumulated straight
//   into the WMMA C fragment (C layout: VGPR r <-> M=r (lanes 0-15) / M=r+8).
//   All hot-loop addressing is 32-bit (saddr+voffset form) and loads are
//   batched into arrays so the compiler can clause them (8 loads in flight).
// ---------------------------------------------------------------------------
__global__ __launch_bounds__(256)
void k_output(const float* __restrict__ WoutT, const int* __restrict__ nodes,
              const float* __restrict__ gvals, float* __restrict__ y) {
    __shared__ float Ash[16][TOPN];   // dense top-of-tree scores (scattered)
    __shared__ int   nd[16][8];       // deep node ids   (depths 4..11)
    __shared__ float gd[16][8];       // deep gelu scores
    int tid = threadIdx.x;
    int s0  = blockIdx.x * 16;

    Ash[tid >> 4][tid & 15] = 0.f;    // 256 threads == 16x16 exactly
    __syncthreads();
    if (tid < 16 * NSTEPS) {          // 192 entries
        int b = tid / NSTEPS, d = tid % NSTEPS;
        int   n  = nodes[(s0 + b) * NSTEPS + d];
        float gv = gvals[(s0 + b) * NSTEPS + d];
        if (d < 4) Ash[b][n] = gv;    // node ids at depth<=3 are 0..14
        else { nd[b][d - 4] = n; gd[b][d - 4] = gv; }
    }
    __syncthreads();

    int wid   = tid >> 5;
    int lane  = tid & 31;
    int hl    = lane & 15;
    int khalf = (lane < 16) ? 0 : 2;  // K offset of this half-wave in A/B frags
    int mbase = (lane < 16) ? 0 : 8;  // M offset of this half-wave in C/D frag

    // A fragments (16x4 per chunk): lane m holds row m, VGPR0=K+khalf, VGPR1=K+khalf+1
    v2f afrag[4];
    #pragma unroll
    for (int k = 0; k < 4; ++k) {
        afrag[k][0] = Ash[hl][4 * k + khalf + 0];
        afrag[k][1] = Ash[hl][4 * k + khalf + 1];
    }

    // Hoist deep metadata (nt-invariant) as 32-bit element offsets + scales.
    int   ro[8][8];
    float gq[8][8];
    #pragma unroll
    for (int r = 0; r < 8; ++r)
        #pragma unroll
        for (int d = 0; d < 8; ++d) {
            ro[r][d] = nd[mbase + r][d] * D_OUT;
            gq[r][d] = gd[mbase + r][d];
        }

    // 2048/16 = 128 N-tiles, 8 waves -> 16 tiles per wave (uniform: EXEC all-1s)
    for (int nt = wid; nt < D_OUT / 16; nt += 8) {
        int ncol = nt * 16;

        // B fragments for all 4 K-chunks, batched into one load burst.
        // B (4x16) chunk k: VGPR0 = rows {4k,4k+2}, VGPR1 = rows {4k+1,4k+3}.
        float bv[8];
        #pragma unroll
        for (int k = 0; k < 4; ++k) {
            bv[2 * k + 0] = WoutT[(4 * k + khalf + 0) * D_OUT + ncol + hl];
            bv[2 * k + 1] = WoutT[(4 * k + khalf + 1) * D_OUT + ncol + hl];
        }

        v8f c = {0.f, 0.f, 0.f, 0.f, 0.f, 0.f, 0.f, 0.f};
        #pragma unroll
        for (int k = 0; k < 4; ++k) {
            v2f b;
            b[0] = bv[2 * k + 0];
            b[1] = bv[2 * k + 1];
            c = __builtin_amdgcn_wmma_f32_16x16x4_f32(false, afrag[k], false, b,
                                                      (short)0, c, false, false);
        }

        // divergent depths 4..11: clause-batched gathers into the C fragment
        int col = ncol + hl;
        #pragma unroll
        for (int r = 0; r < 8; ++r) {
            float wv[8];
            #pragma unroll
            for (int d = 0; d < 8; ++d)
                wv[d] = WoutT[ro[r][d] + col];
            float acc = c[r];
            #pragma unroll
            for (int d = 0; d < 8; ++d)
                acc = fmaf(gq[r][d], wv[d], acc);
            // y is write-once streaming data: keep it out of L2's hot set
            __builtin_nontemporal_store(acc, &y[(s0 + mbase + r) * D_OUT + col]);
        }
    }
}

// ---------------------------------------------------------------------------
extern "C" void kernel_launch(void* const* d_in, const int* in_sizes, int n_in,
                              void* d_out, int out_size, void* d_ws, size_t ws_size,
                              hipStream_t stream) {
    const float* x     = (const float*)d_in[0];   // (8192, 2048)
    const float* W_in  = (const float*)d_in[1];   // (4095, 2048)
    const float* W_out = (const float*)d_in[2];   // (2048, 4095)
    float* y = (float*)d_out;                     // (8192, 2048)

    char* ws = (char*)d_ws;
    size_t offT = (size_t)N_NODES * D_OUT * sizeof(float);          // 33,546,240 B
    float* WoutT = (float*)ws;
    int*   nodes = (int*)(ws + offT);                               // 8192*12 ints
    float* gv    = (float*)(ws + offT + (size_t)B_SAMP * NSTEPS * sizeof(int));

    k_transpose_wout<<<dim3((N_NODES + 31) / 32, D_OUT / 32), dim3(32, 8), 0, stream>>>(W_out, WoutT);
    k_route<<<dim3(B_SAMP / 8), dim3(256), 0, stream>>>(x, W_in, nodes, gv);
    k_output<<<dim3(B_SAMP / 16), dim3(256), 0, stream>>>(WoutT, nodes, gv, y);
}